// ODEINDLayerTestEPS_17609365913956
// MI455X (gfx1250) — compile-verified
//
#include <hip/hip_runtime.h>
#include <hip/hip_bf16.h>
#include <math.h>

// Problem constants (match reference)
#define TT     100
#define O1_    3
#define NV_    300
#define NS_    396
#define NTOT_  696
#define MM     498      // constraint rows
#define MP     512      // padded (16 panels of 32)
#define NBLK   32       // panel / block width
#define NPAN   16       // MP / NBLK
#define BATCH  128

typedef __attribute__((ext_vector_type(2))) float v2f;
typedef __attribute__((ext_vector_type(8))) float v8f;
typedef __attribute__((__vector_size__(4 * sizeof(int)))) int v4i;

typedef __attribute__((address_space(1))) v4i* gv4i_ptr;   // global
typedef __attribute__((address_space(3))) v4i* lv4i_ptr;   // LDS

#if __has_builtin(__builtin_amdgcn_global_load_async_to_lds_b128)
#define HAVE_ASYNC_LDS 1
#else
#define HAVE_ASYNC_LDS 0
#endif

__device__ __forceinline__ float pinv_of(int col) {
    return (col < NV_) ? 100.0f : 0.1f;   // 1/0.01 , 1/10
}

// Build sparse row m of A for batch b. Returns nnz (<=5).
// coeffs: [B, T, 3]; steps: [B, T-1]
__device__ int build_row(int m, int b, const float* __restrict__ coeffs,
                         const float* __restrict__ steps,
                         int* cols, float* vals) {
    if (m < TT) {                      // equation rows
        const float* cr = coeffs + (size_t)b * TT * O1_ + m * O1_;
        cols[0] = m * 3 + 0; vals[0] = cr[0];
        cols[1] = m * 3 + 1; vals[1] = cr[1];
        cols[2] = m * 3 + 2; vals[2] = cr[2];
        return 3;
    }
    if (m < TT + 2) {                  // initial-value rows (NIVS=1, NIV=2)
        int i = m - TT;
        cols[0] = i;                   // (i//2)*3 + (i%2) for i in {0,1}
        vals[0] = 1.0f;
        return 1;
    }
    int s = m - (TT + 2);
    int nn = 0;
    cols[nn] = NV_ + s; vals[nn] = 1.0f; ++nn;   // eps slack
    const bool bwd = (s >= 198);
    const int s2 = bwd ? (s - 198) : s;
    const int t = s2 >> 1;
    const int j = s2 & 1;
    const float h  = steps[(size_t)b * (TT - 1) + t];
    const float h2 = 0.5f * h * h;
    if (!bwd) {
        cols[nn] = (t + 1) * 3 + j; vals[nn] = 1.0f; ++nn;
        if (j == 0) {
            cols[nn] = t * 3 + 0; vals[nn] = -1.0f; ++nn;
            cols[nn] = t * 3 + 1; vals[nn] = -h;    ++nn;
            cols[nn] = t * 3 + 2; vals[nn] = -h2;   ++nn;
        } else {
            cols[nn] = t * 3 + 1; vals[nn] = -1.0f; ++nn;
            cols[nn] = t * 3 + 2; vals[nn] = -h;    ++nn;
        }
    } else {
        cols[nn] = t * 3 + j; vals[nn] = -1.0f; ++nn;
        if (j == 0) {
            cols[nn] = (t + 1) * 3 + 0; vals[nn] = 1.0f; ++nn;
            cols[nn] = (t + 1) * 3 + 1; vals[nn] = -h;   ++nn;
            cols[nn] = (t + 1) * 3 + 2; vals[nn] = h2;   ++nn;
        } else {
            cols[nn] = (t + 1) * 3 + 1; vals[nn] = 1.0f; ++nn;
            cols[nn] = (t + 1) * 3 + 2; vals[nn] = -h;   ++nn;
        }
    }
    return nn;
}

// ---------------------------------------------------------------------------
// Kernel 1: build bvec and dense padded S = A * Pinv * A^T  (one block/batch)
// ---------------------------------------------------------------------------
__global__ void __launch_bounds__(256)
build_system(const float* __restrict__ coeffs, const float* __restrict__ rhs,
             const float* __restrict__ iv_rhs, const float* __restrict__ steps,
             float* __restrict__ S, float* __restrict__ bvec) {
    const int b   = blockIdx.x;
    const int tid = threadIdx.x;
    __shared__ int   scols[MM * 6];
    __shared__ float svals[MM * 6];

    for (int m = tid; m < MM; m += 256) {
        int cols[6]; float vals[6];
        int nn = build_row(m, b, coeffs, steps, cols, vals);
        float bv = 0.0f;
        #pragma unroll
        for (int s = 0; s < 6; ++s) {
            if (s >= nn) { cols[s] = NTOT_; vals[s] = 0.0f; }
            else bv += vals[s] * pinv_of(cols[s]);
            scols[m * 6 + s] = cols[s];
            svals[m * 6 + s] = vals[s];
        }
        if (m < TT)          bv += rhs[(size_t)b * TT + m];
        else if (m < TT + 2) bv += iv_rhs[(size_t)b * 2 + (m - TT)];
        bvec[(size_t)b * MP + m] = bv;
    }
    __syncthreads();

    float* Sb = S + (size_t)b * MP * MP;
    for (int idx = tid; idx < MP * MP; idx += 256) {
        const int i = idx >> 9;
        const int j = idx & (MP - 1);
        float v = 0.0f;
        if (i < MM && j < MM) {
            #pragma unroll
            for (int a = 0; a < 6; ++a) {
                const float va = svals[i * 6 + a];
                if (va == 0.0f) continue;
                const int   ca = scols[i * 6 + a];
                const float pv = pinv_of(ca) * va;
                #pragma unroll
                for (int q = 0; q < 6; ++q)
                    if (scols[j * 6 + q] == ca) v += pv * svals[j * 6 + q];
            }
        } else if (i == j) {
            v = 1.0f;   // SPD padding
        }
        Sb[idx] = v;
    }
}

// ---------------------------------------------------------------------------
// Kernel 2: in-place blocked Cholesky of S (lower), panel width 32.
// Trailing SYRK: per-wave 32x32 register tile = 2x2 accumulators, K=32,
// 32 chained V_WMMA_F32_16X16X4_F32 per tile.
// ---------------------------------------------------------------------------
__global__ void __launch_bounds__(256)
chol_kernel(float* __restrict__ S) {
    const int b     = blockIdx.x;
    float* Sb       = S + (size_t)b * MP * MP;
    const int tid   = threadIdx.x;
    const int lane  = tid & 31;
    const int wave  = tid >> 5;

    __shared__ float pan[MP * NBLK];         // 64 KB column panel, global-row indexed

    for (int k = 0; k < NPAN; ++k) {
        const int k32 = k * NBLK;
        // ---- load panel: rows [k32,512) x cols [k32,k32+32) ----
        {
            const int nch = (MP - k32) * (NBLK / 4);   // 16-byte chunks
#if HAVE_ASYNC_LDS
            for (int ch = tid; ch < nch; ch += 256) {
                const int r  = k32 + (ch >> 3);
                const int c4 = (ch & 7) * 4;
                __builtin_amdgcn_global_load_async_to_lds_b128(
                    (gv4i_ptr)(Sb + (size_t)r * MP + k32 + c4),
                    (lv4i_ptr)(&pan[r * NBLK + c4]),
                    0, 0);
            }
            asm volatile("s_wait_asynccnt 0x0" ::: "memory");
#else
            for (int ch = tid; ch < nch; ch += 256) {
                const int r  = k32 + (ch >> 3);
                const int c4 = (ch & 7) * 4;
                float4 t = *(const float4*)(Sb + (size_t)r * MP + k32 + c4);
                *(float4*)(&pan[r * NBLK + c4]) = t;
            }
#endif
        }
        __syncthreads();

        // ---- factor 32x32 diagonal block (hierarchical, 2x16) ----
        // (a) chol of D00 (16x16), serial
        if (tid == 0) {
            for (int c = 0; c < 16; ++c) {
                float d = pan[(k32 + c) * NBLK + c];
                for (int p = 0; p < c; ++p) {
                    const float l = pan[(k32 + c) * NBLK + p];
                    d -= l * l;
                }
                d = sqrtf(fmaxf(d, 1e-20f));
                pan[(k32 + c) * NBLK + c] = d;
                const float inv = 1.0f / d;
                for (int r = c + 1; r < 16; ++r) {
                    float s = pan[(k32 + r) * NBLK + c];
                    for (int p = 0; p < c; ++p)
                        s -= pan[(k32 + r) * NBLK + p] * pan[(k32 + c) * NBLK + p];
                    pan[(k32 + r) * NBLK + c] = s * inv;
                }
            }
        }
        __syncthreads();
        // (b) TRSM: L10 (rows 16..31 vs D00), one row per thread
        if (tid < 16) {
            const int r = k32 + 16 + tid;
            for (int c = 0; c < 16; ++c) {
                float s = pan[r * NBLK + c];
                for (int p = 0; p < c; ++p)
                    s -= pan[r * NBLK + p] * pan[(k32 + c) * NBLK + p];
                pan[r * NBLK + c] = s / pan[(k32 + c) * NBLK + c];
            }
        }
        __syncthreads();
        // (c) SYRK: D11 -= L10 * L10^T, one element per thread
        {
            const int r = tid >> 4, c = tid & 15;
            float s = pan[(k32 + 16 + r) * NBLK + 16 + c];
            #pragma unroll
            for (int p = 0; p < 16; ++p)
                s -= pan[(k32 + 16 + r) * NBLK + p] * pan[(k32 + 16 + c) * NBLK + p];
            pan[(k32 + 16 + r) * NBLK + 16 + c] = s;
        }
        __syncthreads();
        // (d) chol of D11 (16x16), serial
        if (tid == 0) {
            for (int c = 0; c < 16; ++c) {
                float d = pan[(k32 + 16 + c) * NBLK + 16 + c];
                for (int p = 0; p < c; ++p) {
                    const float l = pan[(k32 + 16 + c) * NBLK + 16 + p];
                    d -= l * l;
                }
                d = sqrtf(fmaxf(d, 1e-20f));
                pan[(k32 + 16 + c) * NBLK + 16 + c] = d;
                const float inv = 1.0f / d;
                for (int r = c + 1; r < 16; ++r) {
                    float s = pan[(k32 + 16 + r) * NBLK + 16 + c];
                    for (int p = 0; p < c; ++p)
                        s -= pan[(k32 + 16 + r) * NBLK + 16 + p] * pan[(k32 + 16 + c) * NBLK + 16 + p];
                    pan[(k32 + 16 + r) * NBLK + 16 + c] = s * inv;
                }
            }
        }
        __syncthreads();

        // ---- TRSM: rows below diag block vs full 32x32 L00 ----
        for (int r = k32 + NBLK + tid; r < MP; r += 256) {
            #pragma unroll 1
            for (int c = 0; c < NBLK; ++c) {
                float s = pan[r * NBLK + c];
                for (int p = 0; p < c; ++p)
                    s -= pan[r * NBLK + p] * pan[(k32 + c) * NBLK + p];
                pan[r * NBLK + c] = s / pan[(k32 + c) * NBLK + c];
            }
        }
        __syncthreads();

        // ---- write panel (L) back to global ----
        for (int idx = tid; idx < (MP - k32) * NBLK; idx += 256) {
            const int r = k32 + (idx >> 5);
            const int c = idx & 31;
            Sb[(size_t)r * MP + k32 + c] = pan[r * NBLK + c];
        }

        // ---- trailing update: 32x32 tiles, wave-uniform dispatch ----
        int tcount = 0;
        for (int bi = k + 1; bi < NPAN; ++bi) {
            for (int bj = k + 1; bj <= bi; ++bj) {
                if ((tcount & 7) == wave) {
                    const int half = lane >> 4;      // 0/1 : C M-half, A/B K-half
                    const int cm   = half * 8;
                    const int cn   = lane & 15;
                    const int mr   = lane & 15;
                    const int koff = half * 2;
                    const int r0 = bi * NBLK, c0 = bj * NBLK;

                    v8f acc00, acc01, acc10, acc11;
                    #pragma unroll
                    for (int v = 0; v < 8; ++v) {
                        acc00[v] = Sb[(size_t)(r0 +      cm + v) * MP + c0 +      cn];
                        acc01[v] = Sb[(size_t)(r0 +      cm + v) * MP + c0 + 16 + cn];
                        acc10[v] = Sb[(size_t)(r0 + 16 + cm + v) * MP + c0 +      cn];
                        acc11[v] = Sb[(size_t)(r0 + 16 + cm + v) * MP + c0 + 16 + cn];
                    }
                    const int ar0 = (r0 +      mr) * NBLK;
                    const int ar1 = (r0 + 16 + mr) * NBLK;
                    const int br0 = (c0 +      mr) * NBLK;
                    const int br1 = (c0 + 16 + mr) * NBLK;
                    #pragma unroll
                    for (int kk = 0; kk < 8; ++kk) {
                        const int kb = kk * 4 + koff;
                        v2f a0, a1, b0, b1;
                        a0.x = -pan[ar0 + kb]; a0.y = -pan[ar0 + kb + 1];
                        a1.x = -pan[ar1 + kb]; a1.y = -pan[ar1 + kb + 1];
                        b0.x =  pan[br0 + kb]; b0.y =  pan[br0 + kb + 1];
                        b1.x =  pan[br1 + kb]; b1.y =  pan[br1 + kb + 1];
                        acc00 = __builtin_amdgcn_wmma_f32_16x16x4_f32(
                                    false, a0, false, b0, (short)0, acc00, false, false);
                        acc01 = __builtin_amdgcn_wmma_f32_16x16x4_f32(
                                    false, a0, false, b1, (short)0, acc01, false, false);
                        acc10 = __builtin_amdgcn_wmma_f32_16x16x4_f32(
                                    false, a1, false, b0, (short)0, acc10, false, false);
                        acc11 = __builtin_amdgcn_wmma_f32_16x16x4_f32(
                                    false, a1, false, b1, (short)0, acc11, false, false);
                    }
                    #pragma unroll
                    for (int v = 0; v < 8; ++v) {
                        Sb[(size_t)(r0 +      cm + v) * MP + c0 +      cn] = acc00[v];
                        Sb[(size_t)(r0 +      cm + v) * MP + c0 + 16 + cn] = acc01[v];
                        Sb[(size_t)(r0 + 16 + cm + v) * MP + c0 +      cn] = acc10[v];
                        Sb[(size_t)(r0 + 16 + cm + v) * MP + c0 + 16 + cn] = acc11[v];
                    }
                }
                ++tcount;
            }
        }
        __threadfence();
        __syncthreads();
    }
}

// ---------------------------------------------------------------------------
// Kernel 3: triangular solves + recover x + scatter outputs (one block/batch)
// ---------------------------------------------------------------------------
__global__ void __launch_bounds__(256)
solve_kernel(const float* __restrict__ coeffs, const float* __restrict__ steps,
             const float* __restrict__ S, const float* __restrict__ bvec,
             float* __restrict__ out) {
    const int b   = blockIdx.x;
    const int tid = threadIdx.x;
    const float* Sb = S + (size_t)b * MP * MP;

    __shared__ float yv[MM];
    __shared__ float lamv[MM];
    __shared__ float acc[NTOT_];

    for (int m = tid; m < MM; m += 256) yv[m] = bvec[(size_t)b * MP + m];
    __syncthreads();

    // forward: L y = b
    for (int c = 0; c < MM; ++c) {
        if (tid == 0) yv[c] = yv[c] / Sb[(size_t)c * MP + c];
        __syncthreads();
        const float yc = yv[c];
        for (int r = c + 1 + tid; r < MM; r += 256) {
            yv[r] -= Sb[(size_t)r * MP + c] * yc;
            __builtin_prefetch(&Sb[(size_t)r * MP + c + 1], 0, 3);  // next column
        }
        __syncthreads();
    }
    // backward: L^T lam = y
    for (int c = MM - 1; c >= 0; --c) {
        if (tid == 0) lamv[c] = yv[c] / Sb[(size_t)c * MP + c];
        __syncthreads();
        const float lc = lamv[c];
        for (int r = tid; r < c; r += 256)
            yv[r] -= Sb[(size_t)c * MP + r] * lc;   // L^T[r][c] = L[c][r]
        __syncthreads();
    }

    // x = -Pinv * (A^T lam - 1): sparse scatter of A^T lam into LDS
    for (int n = tid; n < NTOT_; n += 256) acc[n] = 0.0f;
    __syncthreads();
    for (int m = tid; m < MM; m += 256) {
        int cols[6]; float vals[6];
        const int nn = build_row(m, b, coeffs, steps, cols, vals);
        const float lm = lamv[m];
        for (int s = 0; s < nn; ++s)
            atomicAdd(&acc[cols[s]], vals[s] * lm);   // ds_add_f32
    }
    __syncthreads();

    float* u0  = out;                 // [B,T] each (B = BS*NID flattened)
    float* u1  = out + 12800;
    float* u2  = out + 25600;
    float* eps = out + 38400;         // [B, NS]
    float* st  = out + 89088;         // [B, T-1]
    float* lo  = out + 101760;        // [B, M]

    for (int n = tid; n < NTOT_; n += 256) {
        const float pv = pinv_of(n);
        const float xv = -pv * (acc[n] - 1.0f);
        if (n < NV_) {
            const int t = n / 3, j = n % 3;
            float* dst = (j == 0) ? u0 : ((j == 1) ? u1 : u2);
            dst[(size_t)b * TT + t] = xv;
        } else {
            eps[(size_t)b * NS_ + (n - NV_)] = fabsf(xv);
        }
    }
    for (int m = tid; m < MM; m += 256) lo[(size_t)b * MM + m] = lamv[m];
    for (int t = tid; t < TT - 1; t += 256)
        st[(size_t)b * (TT - 1) + t] = steps[(size_t)b * (TT - 1) + t];
}

// ---------------------------------------------------------------------------
extern "C" void kernel_launch(void* const* d_in, const int* in_sizes, int n_in,
                              void* d_out, int out_size, void* d_ws, size_t ws_size,
                              hipStream_t stream) {
    const float* coeffs = (const float*)d_in[0];   // [2,64,100,3]
    const float* rhs    = (const float*)d_in[1];   // [2,64,100]
    const float* iv_rhs = (const float*)d_in[2];   // [2,64,2]
    const float* steps  = (const float*)d_in[3];   // [2,64,99]
    // d_in[4] = lam_init (ignored by the direct KKT solve, as in reference)

    float* S    = (float*)d_ws;                      // 128 * 512*512 f32 = 128 MB
    float* bvec = S + (size_t)BATCH * MP * MP;       // 128 * 512 f32
    float* out  = (float*)d_out;

    build_system<<<dim3(BATCH), dim3(256), 0, stream>>>(coeffs, rhs, iv_rhs, steps, S, bvec);
    chol_kernel <<<dim3(BATCH), dim3(256), 0, stream>>>(S);
    solve_kernel<<<dim3(BATCH), dim3(256), 0, stream>>>(coeffs, steps, S, bvec, out);
}